// PILayer_83313775608061
// MI455X (gfx1250) — compile-verified
//
#include <hip/hip_runtime.h>
#include <math.h>

#define PP 1024
#define LL 32768
#define UU 64
#define KSPLIT 8
#define SEG (LL / KSPLIT)      // 4096
#define CK 256                 // k-chunk per iteration in fused kernel
#define NCHUNK (SEG / CK)      // 16
#define MT 16                  // row (play) tile
#define XPAD 4                 // LDS row padding (keeps 16B align, breaks bank conflicts)

typedef __attribute__((ext_vector_type(2))) float v2f;
typedef __attribute__((ext_vector_type(4))) float v4f;
typedef __attribute__((ext_vector_type(8))) float v8f;
typedef __attribute__((ext_vector_type(4))) int   v4i;

// CDNA5 async global->LDS copy (ASYNCcnt-tracked), guarded so either toolchain builds.
#if defined(__HIP_DEVICE_COMPILE__) && __has_builtin(__builtin_amdgcn_global_load_async_to_lds_b128)
#define HAVE_ASYNC_LDS 1
#else
#define HAVE_ASYNC_LDS 0
#endif

#if HAVE_ASYNC_LDS
// builtin signature (from probe): (v4i addrspace(1)*, v4i addrspace(3)*, imm i32 offset, imm i32 cpol)
typedef __attribute__((address_space(1))) v4i* gptr_v4i;
typedef __attribute__((address_space(3))) v4i* lptr_v4i;
__device__ __forceinline__ void async_cp16(float* lds_dst, const float* gsrc) {
    __builtin_amdgcn_global_load_async_to_lds_b128(
        (gptr_v4i)(gsrc), (lptr_v4i)(lds_dst), 0, 0);
}
#endif

__device__ __forceinline__ void wait_async0() {
#if defined(__HIP_DEVICE_COMPILE__)
#if __has_builtin(__builtin_amdgcn_s_wait_asynccnt)
    __builtin_amdgcn_s_wait_asynccnt(0);
#else
    asm volatile("s_wait_asynccnt 0" ::: "memory");
#endif
#endif
}

// Compose running clamp (a,b) with step clamp (z, z+1): g' = step ∘ g
__device__ __forceinline__ void comp(float& a, float& b, float z) {
    float zh = z + 1.0f;
    a = fminf(fmaxf(a, z), zh);
    b = fminf(fmaxf(b, z), zh);
}

// ---------------------------------------------------------------------------
// Kernel 1: per (play, split) composed clamp summary. One wave per summary.
// Lane j sequentially composes 128 contiguous elements (float4 loads), then an
// ordered shuffle-tree reduction (offsets 1,2,4,8,16) merges lanes in order.
// ---------------------------------------------------------------------------
__global__ __launch_bounds__(128) void seg_summary(const float* __restrict__ x,
                                                   const float* __restrict__ pw,
                                                   float2* __restrict__ sum) {
    const int wave = (blockIdx.x * blockDim.x + threadIdx.x) >> 5;
    const int lane = threadIdx.x & 31;
    const int p = wave / KSPLIT;
    const int s = wave % KSPLIT;
    const float w = pw[p];
    const int LPL = SEG / 32;  // 128 elements per lane
    const float* xr = x + (size_t)p * LL + (size_t)s * SEG + (size_t)lane * LPL;
    const int jbase0 = s * SEG + lane * LPL;

    float a = -INFINITY, b = INFINITY;
    {   // first float4: global column 0 (== p0, not a recurrence step) is skipped
        v4f v = *(const v4f*)xr;
        if (jbase0 == 0) {
            comp(a, b, v[1] * w); comp(a, b, v[2] * w); comp(a, b, v[3] * w);
        } else {
            comp(a, b, v[0] * w); comp(a, b, v[1] * w);
            comp(a, b, v[2] * w); comp(a, b, v[3] * w);
        }
    }
    for (int i = 1; i < LPL / 4; ++i) {
        v4f v = *(const v4f*)(xr + i * 4);
#pragma unroll
        for (int e = 0; e < 4; ++e) comp(a, b, v[e] * w);
    }
    // ordered reduction: after offset o, lane i holds range [i, i+2o)
#pragma unroll
    for (int off = 1; off < 32; off <<= 1) {
        float oa = __shfl_down(a, off, 32);
        float ob = __shfl_down(b, off, 32);
        float na = fminf(fmaxf(a, oa), ob);
        float nb = fminf(fmaxf(b, oa), ob);
        a = na; b = nb;
    }
    if (lane == 0) sum[wave] = make_float2(a, b);
}

// ---------------------------------------------------------------------------
// Kernel 2: fused scan + GEMM. Block = 128 threads (4 waves).
// blockIdx.x = row tile (16 plays), blockIdx.y = k-split.
// Double-buffered x staging (async-to-LDS when available); each wave owns 16
// of the 64 output columns and issues v_wmma_f32_16x16x4_f32.
// ---------------------------------------------------------------------------
__global__ __launch_bounds__(128) void pi_fused(const float* __restrict__ x,
                                                const float* __restrict__ pw,
                                                const float* __restrict__ kern,
                                                const float2* __restrict__ sum,
                                                float* __restrict__ part) {
    __shared__ __align__(16) float xs[2][MT][CK + XPAD];
    __shared__ __align__(16) float ys[MT][CK + XPAD];
    __shared__ float seg_a[MT][8], seg_b[MT][8];
    __shared__ float ex_a[MT][8], ex_b[MT][8];
    __shared__ float carry_a[MT], carry_b[MT], p0s[MT], wrow[MT];

    const int t = threadIdx.x;
    const int lane = t & 31;
    const int wv = t >> 5;           // wave id -> N tile (16 cols)
    const int rt = blockIdx.x;       // row tile
    const int s = blockIdx.y;        // k-split
    const int r = t & 15;            // scan: row
    const int ss = t >> 4;           // scan: subsegment 0..7 (32 k each)
    const int halfsel = lane >> 4;   // WMMA half: K pair {0,1} vs {2,3}
    const int ncol = lane & 15;
    const int rbase = t >> 6;        // chunk-load: row parity
    const int c4 = t & 63;           // chunk-load: float4 column

    const float* xsplit = x + (size_t)(rt * MT) * LL + (size_t)s * SEG;

    // carry-in for this split: compose summaries of splits 0..s-1, per row
    if (t < MT) {
        const int p = rt * MT + t;
        float ca = -INFINITY, cb = INFINITY;
        for (int s2 = 0; s2 < s; ++s2) {
            float2 sv = sum[p * KSPLIT + s2];
            ca = fminf(fmaxf(ca, sv.x), sv.y);
            cb = fminf(fmaxf(cb, sv.x), sv.y);
        }
        carry_a[t] = ca; carry_b[t] = cb;
        p0s[t] = x[(size_t)p * LL];
        wrow[t] = pw[p];
    }

    // prologue: stage chunk 0 into xs[0]
#if HAVE_ASYNC_LDS
#pragma unroll
    for (int i = 0; i < 8; ++i) {
        const int row = 2 * i + rbase;
        async_cp16(&xs[0][row][c4 * 4], xsplit + (size_t)row * LL + c4 * 4);
    }
    wait_async0();
#else
#pragma unroll
    for (int i = 0; i < 8; ++i) {
        const int row = 2 * i + rbase;
        *(v4f*)&xs[0][row][c4 * 4] = *(const v4f*)(xsplit + (size_t)row * LL + c4 * 4);
    }
#endif
    __syncthreads();

    v8f acc = {0.f, 0.f, 0.f, 0.f, 0.f, 0.f, 0.f, 0.f};
#if !HAVE_ASYNC_LDS
    v4f stage[8];
#endif

    for (int ch = 0; ch < NCHUNK; ++ch) {
        const int cb = ch & 1, nb = cb ^ 1;
        const int k0 = s * SEG + ch * CK;  // global column base of this chunk
        const bool havenext = (ch + 1 < NCHUNK);

        // ---- issue next chunk's staging early (hidden behind scan + GEMM) ----
        if (havenext) {
            const float* gnext = xsplit + (size_t)(ch + 1) * CK;
#if HAVE_ASYNC_LDS
#pragma unroll
            for (int i = 0; i < 8; ++i) {
                const int row = 2 * i + rbase;
                async_cp16(&xs[nb][row][c4 * 4], gnext + (size_t)row * LL + c4 * 4);
            }
#else
#pragma unroll
            for (int i = 0; i < 8; ++i) {
                const int row = 2 * i + rbase;
                stage[i] = *(const v4f*)(gnext + (size_t)row * LL + c4 * 4);
            }
#endif
            // prime L2 with next chunk's B tile
            __builtin_prefetch(kern + (size_t)(k0 + CK) * UU + t * 128, 0, 1);
        }

        // ---- pass A: subsegment clamp summaries (32 elements each) ----
        {
            const float w = wrow[r];
            const int base = ss * 32;
            float a = -INFINITY, b = INFINITY;
            if (k0 + base == 0) {          // subsegment containing global j==0
                for (int c = 1; c < 32; ++c) comp(a, b, xs[cb][r][base + c] * w);
            } else {
#pragma unroll 8
                for (int c = 0; c < 32; ++c) comp(a, b, xs[cb][r][base + c] * w);
            }
            seg_a[r][ss] = a; seg_b[r][ss] = b;
        }
        __syncthreads();

        // ---- carry scan over 8 subsegments (16 threads, 8 fixed steps) ----
        if (t < MT) {
            float ca = carry_a[t], cb2 = carry_b[t];
#pragma unroll
            for (int q = 0; q < 8; ++q) {
                ex_a[t][q] = ca; ex_b[t][q] = cb2;
                float sa = seg_a[t][q], sb = seg_b[t][q];
                ca = fminf(fmaxf(ca, sa), sb);
                cb2 = fminf(fmaxf(cb2, sa), sb);
            }
            carry_a[t] = ca; carry_b[t] = cb2;  // carry into next chunk
        }
        __syncthreads();

        // ---- pass B: reconstruct y_j = clamp(p0, prefix) into LDS ----
        {
            float pa = ex_a[r][ss], pb = ex_b[r][ss];
            const float w = wrow[r], p0 = p0s[r];
            const int base = ss * 32;
            if (k0 + base == 0) {
                ys[r][0] = fminf(fmaxf(p0, pa), pb);   // y_0 = p0 (identity prefix)
                for (int c = 1; c < 32; ++c) {
                    comp(pa, pb, xs[cb][r][base + c] * w);
                    ys[r][base + c] = fminf(fmaxf(p0, pa), pb);
                }
            } else {
#pragma unroll 8
                for (int c = 0; c < 32; ++c) {
                    comp(pa, pb, xs[cb][r][base + c] * w);
                    ys[r][base + c] = fminf(fmaxf(p0, pa), pb);
                }
            }
        }
#if !HAVE_ASYNC_LDS
        if (havenext) {   // publish staged next chunk (latency already hidden)
#pragma unroll
            for (int i = 0; i < 8; ++i) {
                const int row = 2 * i + rbase;
                *(v4f*)&xs[nb][row][c4 * 4] = stage[i];
            }
        }
#endif
        __syncthreads();

        // ---- GEMM: 64 x v_wmma_f32_16x16x4_f32 per wave per chunk ----
        {
            const float* kb = kern + (size_t)k0 * UU + wv * 16 + ncol;
#pragma unroll 4
            for (int kk = 0; kk < CK / 4; ++kk) {
                const int kc = kk * 4 + halfsel * 2;
                v2f a2, b2;
                a2.x = ys[ncol][kc];            // A: M=ncol, K=kc
                a2.y = ys[ncol][kc + 1];
                b2.x = kb[(size_t)kc * UU];     // B: K=kc, N=col
                b2.y = kb[(size_t)(kc + 1) * UU];
                acc = __builtin_amdgcn_wmma_f32_16x16x4_f32(
                    false, a2, false, b2, (short)0, acc, false, false);
            }
        }
#if HAVE_ASYNC_LDS
        wait_async0();   // xs[nb] complete before next iteration's pass A
#endif
        __syncthreads();
    }

    // ---- write partial 16x16 tile (C layout: vgpr v -> M = (lane>=16)*8+v) ----
    {
        float* pb = part + ((size_t)s * PP + (size_t)rt * MT) * UU;
        const int n = wv * 16 + ncol;
        const int mbase = halfsel * 8;
#pragma unroll
        for (int v = 0; v < 8; ++v) {
            pb[(size_t)(mbase + v) * UU + n] = acc[v];
        }
    }
}

// ---------------------------------------------------------------------------
// Kernel 3: fixed-order reduction of 8 partials + bias, tanh.
// ---------------------------------------------------------------------------
__global__ __launch_bounds__(256) void finalize(const float* __restrict__ part,
                                                const float* __restrict__ bias,
                                                float* __restrict__ out) {
    const int i = blockIdx.x * blockDim.x + threadIdx.x;
    if (i >= PP * UU) return;
    float a = bias[i & (UU - 1)];
#pragma unroll
    for (int s = 0; s < KSPLIT; ++s) a += part[(size_t)s * PP * UU + i];
    out[i] = tanhf(a);
}

extern "C" void kernel_launch(void* const* d_in, const int* in_sizes, int n_in,
                              void* d_out, int out_size, void* d_ws, size_t ws_size,
                              hipStream_t stream) {
    const float* x = (const float*)d_in[0];
    const float* pw = (const float*)d_in[1];
    const float* kern = (const float*)d_in[2];
    const float* bias = (const float*)d_in[3];
    float* out = (float*)d_out;

    float2* sum = (float2*)d_ws;                            // 1024*8*8B = 64 KB
    float* part = (float*)((char*)d_ws + 64 * 1024);        // 8*1024*64*4B = 2 MB

    seg_summary<<<dim3((PP * KSPLIT * 32) / 128), dim3(128), 0, stream>>>(x, pw, sum);
    pi_fused<<<dim3(PP / MT, KSPLIT), dim3(128), 0, stream>>>(x, pw, kern, sum, part);
    finalize<<<dim3((PP * UU) / 256), dim3(256), 0, stream>>>(part, bias, out);
}